// PGA_OPTIMIZOR_TORCH_64364379898439
// MI455X (gfx1250) — compile-verified
//
#include <hip/hip_runtime.h>
#include <math.h>

#define NPTS 4096
#define NV   (NPTS / 2)        // 2048 float4 entries (2 points per float4)
#define NSEG 4                 // column segments per row
#define SEGV (NV / NSEG)       // 512 float4s per segment
#define NBLK (8 * NSEG)        // 32 blocks

#define R_EARTH 6373.0f

// ---------------------------------------------------------------------------
// Pass 1: per-point haversine + log transform.  x_i = log(pga_i) + c * d_i
// ---------------------------------------------------------------------------
__global__ void prep_kernel(const float* __restrict__ mat,
                            const float* __restrict__ lat,
                            const float* __restrict__ lon,
                            const float* __restrict__ c,
                            float2* __restrict__ xp) {
    int i = blockIdx.x * blockDim.x + threadIdx.x;
    if (i >= NPTS) return;
    const float D2R = (float)(3.14 / 180.0);
    float lat1 = lat[0] * D2R;
    float lon1 = lon[0] * D2R;
    float lat2 = mat[i * 3 + 0] * D2R;
    float lon2 = mat[i * 3 + 1] * D2R;
    float pga  = mat[i * 3 + 2];
    float dlat = lat2 - lat1;
    float dlon = lon2 - lon1;
    float sdlat = sinf(dlat * 0.5f);
    float sdlon = sinf(dlon * 0.5f);
    float a = sdlat * sdlat + cosf(lat1) * cosf(lat2) * sdlon * sdlon;
    float d = R_EARTH * 2.0f * atan2f(sqrtf(a), sqrtf(1.0f - a));
    float x = logf(pga) + c[0] * d;
    xp[i] = make_float2(x, pga);
}

// ---------------------------------------------------------------------------
// Pass 2: pairwise upper-triangle reduction.
// 32 blocks: blockIdx[2:0] selects a 256-row group (thread t handles rows t
// and 4095-t -> balanced), blockIdx[4:3] selects a 512-float4 column segment.
// Each block async-DMAs the full 32KB (x,p) table into LDS, then streams the
// segment with float4 (ds_load_b128) reads; fp32 accumulation per segment,
// flushed to f64 once per row.
// ---------------------------------------------------------------------------
__global__ __launch_bounds__(256) void pair_kernel(const float4* __restrict__ xp4,
                                                   double* __restrict__ block_sums) {
    __shared__ float4 s_v4[NV];    // 32 KB
    __shared__ double s_red[256];  //  2 KB

    const int tid = threadIdx.x;

    // --- CDNA5 async global->LDS staging: 256 threads x 8 x 16B = 32KB ---
    unsigned int lds_base = (unsigned int)(unsigned long long)(&s_v4[0]);
    unsigned long long gbase = (unsigned long long)xp4;
#pragma unroll
    for (int k = 0; k < 8; ++k) {
        unsigned int off = (((unsigned int)tid) + (((unsigned int)k) << 8)) << 4;
        asm volatile("global_load_async_to_lds_b128 %0, %1, off"
                     :: "v"(lds_base + off), "v"(gbase + off)
                     : "memory");
    }
    asm volatile("s_wait_asynccnt 0" ::: "memory");
    __syncthreads();

    const float2* s_xp = (const float2*)s_v4;

    const int seg  = blockIdx.x >> 3;                 // 0..3
    const int t    = ((blockIdx.x & 7) << 8) | tid;   // 0..2047
    const int v_lo = seg * SEGV;
    const int v_hi = v_lo + SEGV;

    double acc = 0.0;

#pragma unroll
    for (int r = 0; r < 2; ++r) {
        const int i = r ? (NPTS - 1 - t) : t;
        const float xi = s_xp[i].x;
        const float pi = s_xp[i].y;

        // scalar peel: first column j=i+1 if it is the odd half of a float4
        float facc = 0.0f;
        const int j1 = i + 1;
        if ((j1 < NPTS) && (j1 & 1)) {
            const int vj = j1 >> 1;
            if (vj >= v_lo && vj < v_hi) {
                float2 o = s_xp[j1];
                facc = (xi - o.x) * fminf(pi, o.y);
            }
        }

        // vector body over this block's column segment
        int jv = (i + 2) >> 1;
        jv = (jv > v_lo) ? jv : v_lo;
        float f0 = 0.0f, f1 = 0.0f;
        for (; jv < v_hi; ++jv) {
            float4 q = s_v4[jv];
            f0 += (xi - q.x) * fminf(pi, q.y);
            f1 += (xi - q.z) * fminf(pi, q.w);
        }
        acc += (double)(facc + f0 + f1);   // <=1025 fp32 terms per flush
    }

    // --- block reduction (double) ---
    s_red[tid] = acc;
    __syncthreads();
#pragma unroll
    for (int s = 128; s > 0; s >>= 1) {
        if (tid < s) s_red[tid] += s_red[tid + s];
        __syncthreads();
    }
    if (tid == 0) block_sums[blockIdx.x] = s_red[0];
}

// ---------------------------------------------------------------------------
// Pass 3: combine block sums, apply the c<0 penalty term, cast to fp32.
// ---------------------------------------------------------------------------
__global__ void final_kernel(const double* __restrict__ block_sums,
                             const float* __restrict__ c,
                             float* __restrict__ out) {
    double s = 0.0;
#pragma unroll
    for (int b = 0; b < NBLK; ++b) s += block_sums[b];
    float cv = c[0];
    if (cv < 0.0f) s -= (double)(10.0f * cv);
    out[0] = (float)s;
}

// ---------------------------------------------------------------------------
extern "C" void kernel_launch(void* const* d_in, const int* in_sizes, int n_in,
                              void* d_out, int out_size, void* d_ws, size_t ws_size,
                              hipStream_t stream) {
    const float* mat = (const float*)d_in[0];
    const float* lat = (const float*)d_in[1];
    const float* lon = (const float*)d_in[2];
    const float* c   = (const float*)d_in[3];

    double* block_sums = (double*)d_ws;                    // 32 doubles = 256 B
    float2* xp         = (float2*)((char*)d_ws + 256);     // 32 KB, 16B-aligned

    prep_kernel <<<  16, 256, 0, stream>>>(mat, lat, lon, c, xp);
    pair_kernel <<<NBLK, 256, 0, stream>>>((const float4*)xp, block_sums);
    final_kernel<<<   1,   1, 0, stream>>>(block_sums, c, (float*)d_out);
}